// Mamba_71451075937043
// MI455X (gfx1250) — compile-verified
//
#include <hip/hip_runtime.h>
#include <math.h>

// ---------------- model dims ----------------
#define D_MODEL 1024
#define ED      2048
#define SEQ     2048
#define DT_RANK 64
#define D_STATE 16
#define D_CONV  4
#define N_LAYERS 2
#define EPS     1e-5f

// scan chunking
#define CHUNKS  16
#define LCHUNK  (SEQ / CHUNKS)   // 128
#define DBC_LD  (DT_RANK + 2 * D_STATE)   // 96

typedef __attribute__((ext_vector_type(2))) float v2f;
typedef __attribute__((ext_vector_type(8))) float v8f;
typedef __attribute__((ext_vector_type(4))) int   v4i;

// ---------------- CDNA5 async global->LDS copy helpers ----------------
#if __has_builtin(__builtin_amdgcn_global_load_async_to_lds_b128)
#define HAS_ASYNC_LDS 1
#else
#define HAS_ASYNC_LDS 0
#endif

typedef __attribute__((address_space(1))) v4i glob_v4i;
typedef __attribute__((address_space(3))) v4i lds_v4i;

__device__ __forceinline__ void async_copy16(const float* __restrict__ g,
                                             float* __restrict__ l) {
#if HAS_ASYNC_LDS
    __builtin_amdgcn_global_load_async_to_lds_b128(
        (glob_v4i*)g, (lds_v4i*)l, 0, 0);
#else
    float4 t = *(const float4*)g;
    *(float4*)l = t;
#endif
}

__device__ __forceinline__ void wait_async_lds() {
#if HAS_ASYNC_LDS
#if __has_builtin(__builtin_amdgcn_s_wait_asynccnt)
    __builtin_amdgcn_s_wait_asynccnt(0);
#else
    asm volatile("s_wait_asynccnt 0x0" ::: "memory");
#endif
#endif
}

// =============================================================
// RMSNorm: one block per row (L rows), 256 threads
// =============================================================
__global__ void rmsnorm_kernel(const float* __restrict__ x,
                               const float* __restrict__ w,
                               float* __restrict__ out) {
    const int l = blockIdx.x;
    const float* row = x + (size_t)l * D_MODEL;
    float s = 0.f;
    for (int i = threadIdx.x; i < D_MODEL; i += 256) {
        float v = row[i];
        s += v * v;
    }
    for (int off = 16; off > 0; off >>= 1) s += __shfl_down(s, off, 32);
    __shared__ float red[8];
    __shared__ float scale_sh;
    if ((threadIdx.x & 31) == 0) red[threadIdx.x >> 5] = s;
    __syncthreads();
    if (threadIdx.x == 0) {
        float tot = 0.f;
        #pragma unroll
        for (int i = 0; i < 8; ++i) tot += red[i];
        scale_sh = rsqrtf(tot / (float)D_MODEL + EPS);
    }
    __syncthreads();
    const float sc = scale_sh;
    for (int i = threadIdx.x; i < D_MODEL; i += 256)
        out[(size_t)l * D_MODEL + i] = row[i] * sc * w[i];
}

// =============================================================
// Generic fp32 WMMA GEMM: C[M][N] = A[M][K] * Bw[N][K]^T  (+epilogue)
// One wave per block (blockDim = 32).
// Wave computes a (16*MT) x (16*NT) tile: MT*NT accumulators, so each
// k4-step issues (MT+NT) float2 loads feeding MT*NT v_wmma ops.
// EPI: 0 = none, 1 = softplus(acc + bias[n]), 2 = acc + res[m*ldc+n]
// =============================================================
template<int MT, int NT, int EPI>
__global__ void gemm_wmma_kernel(const float* __restrict__ A, int lda,
                                 const float* __restrict__ Bw, int ldb,
                                 float* __restrict__ C, int ldc,
                                 int K,
                                 const float* __restrict__ bias,
                                 const float* __restrict__ res) {
    const int t      = threadIdx.x;      // 0..31
    const int lane16 = t & 15;
    const int half   = t >> 4;
    const int m0 = blockIdx.y * (16 * MT);
    const int n0 = blockIdx.x * (16 * NT);

    const float* Arow[MT];
    #pragma unroll
    for (int i = 0; i < MT; ++i)
        Arow[i] = A + (size_t)(m0 + i * 16 + lane16) * lda + half * 2;
    const float* Brow[NT];
    #pragma unroll
    for (int j = 0; j < NT; ++j)
        Brow[j] = Bw + (size_t)(n0 + j * 16 + lane16) * ldb + half * 2;

    v8f acc[MT][NT] = {};

    for (int k = 0; k < K; k += 4) {
        v2f a[MT];
        #pragma unroll
        for (int i = 0; i < MT; ++i) a[i] = *(const v2f*)(Arow[i] + k);
        v2f b[NT];
        #pragma unroll
        for (int j = 0; j < NT; ++j) b[j] = *(const v2f*)(Brow[j] + k);
        #pragma unroll
        for (int i = 0; i < MT; ++i)
            #pragma unroll
            for (int j = 0; j < NT; ++j)
                acc[i][j] = __builtin_amdgcn_wmma_f32_16x16x4_f32(
                    false, a[i], false, b[j], (short)0, acc[i][j], false, false);
    }

    // epilogue + store: acc VGPR v -> row m0+i*16 + v + 8*half, lane -> col
    #pragma unroll
    for (int i = 0; i < MT; ++i) {
        #pragma unroll
        for (int j = 0; j < NT; ++j) {
            const int n = n0 + j * 16 + lane16;
            #pragma unroll
            for (int v = 0; v < 8; ++v) {
                const int m = m0 + i * 16 + v + half * 8;
                float val = acc[i][j][v];
                if (EPI == 1) {
                    float xv = val + bias[n];
                    val = (xv > 20.f) ? xv : log1pf(__expf(xv));
                } else if (EPI == 2) {
                    val += res[(size_t)m * ldc + n];
                }
                C[(size_t)m * ldc + n] = val;
            }
        }
    }
}

// =============================================================
// Causal depthwise conv (K=4) + SiLU.  in = first half of xz rows.
// =============================================================
__global__ void conv_silu_kernel(const float* __restrict__ xz,
                                 const float* __restrict__ cw,
                                 const float* __restrict__ cb,
                                 float* __restrict__ xc) {
    const int idx = blockIdx.x * blockDim.x + threadIdx.x;  // l*ED + e
    if (idx >= SEQ * ED) return;
    const int e = idx & (ED - 1);
    const int l = idx >> 11;
    float acc = cb[e];
    #pragma unroll
    for (int j = 0; j < D_CONV; ++j) {
        const int ll = l - (D_CONV - 1) + j;
        if (ll >= 0) acc += cw[e * D_CONV + j] * xz[(size_t)ll * (2 * ED) + e];
    }
    xc[idx] = acc / (1.f + __expf(-acc));   // silu
}

// =============================================================
// Selective scan, pass 1: per (chunk c, channel e) compute
//   aprod[n] = prod a_t ,  bend[n] = h at chunk end assuming h_in = 0
// The whole block shares one chunk; its B rows are staged into LDS
// with async global->LDS copies (ASYNCcnt path).
// =============================================================
__global__ void scan_pass1(const float* __restrict__ delta,
                           const float* __restrict__ xc,
                           const float* __restrict__ dBC,
                           const float* __restrict__ A_log,
                           float* __restrict__ Aprod,
                           float* __restrict__ Bend) {
    const int idx = blockIdx.x * blockDim.x + threadIdx.x;  // c*ED + e
    const int e = idx & (ED - 1);
    const int c = idx >> 11;         // constant within the block
    const int l0 = c * LCHUNK;

    // stage B rows of this chunk: LCHUNK x 16 floats (8 KB)
    __shared__ float Bs[LCHUNK][D_STATE];
    for (int s = threadIdx.x; s < LCHUNK * 4; s += 256) {
        const int row = s >> 2, seg = s & 3;
        async_copy16(dBC + (size_t)(l0 + row) * DBC_LD + DT_RANK + seg * 4,
                     &Bs[row][seg * 4]);
    }
    wait_async_lds();
    __syncthreads();

    float Ae[D_STATE], ap[D_STATE], h[D_STATE];
    #pragma unroll
    for (int n = 0; n < D_STATE; ++n) {
        Ae[n] = -__expf(A_log[e * D_STATE + n]);
        ap[n] = 1.f;
        h[n]  = 0.f;
    }
    for (int i = 0; i < LCHUNK; ++i) {
        const int l = l0 + i;
        const float d  = delta[(size_t)l * ED + e];
        const float u  = xc[(size_t)l * ED + e];
        const float du = d * u;
        #pragma unroll
        for (int n = 0; n < D_STATE; ++n) {
            const float a = __expf(d * Ae[n]);
            h[n]  = a * h[n] + du * Bs[i][n];
            ap[n] *= a;
        }
    }
    const size_t base = (size_t)idx * D_STATE;
    #pragma unroll
    for (int n = 0; n < D_STATE; ++n) {
        Aprod[base + n] = ap[n];
        Bend [base + n] = h[n];
    }
}

// =============================================================
// pass 2: carry propagation across chunks, thread per (e,n)
// =============================================================
__global__ void scan_pass2(const float* __restrict__ Aprod,
                           const float* __restrict__ Bend,
                           float* __restrict__ Hin) {
    const int idx = blockIdx.x * blockDim.x + threadIdx.x;  // e*16 + n
    if (idx >= ED * D_STATE) return;
    float h = 0.f;
    #pragma unroll
    for (int c = 0; c < CHUNKS; ++c) {
        const size_t o = (size_t)c * (ED * D_STATE) + idx;
        Hin[o] = h;
        h = Aprod[o] * h + Bend[o];
    }
}

// =============================================================
// pass 3: replay chunk with correct h_in; fuse C-contraction,
//         D skip-connection and SiLU(z) gate -> yg.
// B and C rows staged into LDS via async copies.
// =============================================================
__global__ void scan_pass3(const float* __restrict__ delta,
                           const float* __restrict__ xc,
                           const float* __restrict__ dBC,
                           const float* __restrict__ A_log,
                           const float* __restrict__ Dp,
                           const float* __restrict__ xz,
                           const float* __restrict__ Hin,
                           float* __restrict__ yg) {
    const int idx = blockIdx.x * blockDim.x + threadIdx.x;  // c*ED + e
    const int e = idx & (ED - 1);
    const int c = idx >> 11;
    const int l0 = c * LCHUNK;

    // stage B (cols 0..15) and C (cols 16..31) rows: LCHUNK x 32 floats (16 KB)
    __shared__ float BCs[LCHUNK][2 * D_STATE];
    for (int s = threadIdx.x; s < LCHUNK * 8; s += 256) {
        const int row = s >> 3, seg = s & 7;
        async_copy16(dBC + (size_t)(l0 + row) * DBC_LD + DT_RANK + seg * 4,
                     &BCs[row][seg * 4]);
    }
    wait_async_lds();
    __syncthreads();

    float Ae[D_STATE], h[D_STATE];
    const size_t hbase = (size_t)c * (ED * D_STATE) + (size_t)e * D_STATE;
    #pragma unroll
    for (int n = 0; n < D_STATE; ++n) {
        Ae[n] = -__expf(A_log[e * D_STATE + n]);
        h[n]  = Hin[hbase + n];
    }
    const float dskip = Dp[e];
    for (int i = 0; i < LCHUNK; ++i) {
        const int l = l0 + i;
        const float d  = delta[(size_t)l * ED + e];
        const float u  = xc[(size_t)l * ED + e];
        const float du = d * u;
        float y = 0.f;
        #pragma unroll
        for (int n = 0; n < D_STATE; ++n) {
            const float a = __expf(d * Ae[n]);
            h[n] = a * h[n] + du * BCs[i][n];
            y += h[n] * BCs[i][D_STATE + n];
        }
        y += dskip * u;
        const float z = xz[(size_t)l * (2 * ED) + ED + e];
        yg[(size_t)l * ED + e] = y * (z / (1.f + __expf(-z)));
    }
}

// =============================================================
// host launcher
// =============================================================
extern "C" void kernel_launch(void* const* d_in, const int* in_sizes, int n_in,
                              void* d_out, int out_size, void* d_ws, size_t ws_size,
                              hipStream_t stream) {
    (void)in_sizes; (void)n_in; (void)out_size; (void)ws_size;

    const float* x_in   = (const float*)d_in[0];
    const float* norm_w = (const float*)d_in[1];
    const float* Win    = (const float*)d_in[2];
    const float* conv_w = (const float*)d_in[3];
    const float* conv_b = (const float*)d_in[4];
    const float* Wx     = (const float*)d_in[5];
    const float* Wdt    = (const float*)d_in[6];
    const float* bdt    = (const float*)d_in[7];
    const float* A_log  = (const float*)d_in[8];
    const float* Dp     = (const float*)d_in[9];
    const float* Wout   = (const float*)d_in[10];
    float* out = (float*)d_out;

    // workspace layout (floats)
    float* W = (float*)d_ws;
    size_t o = 0;
    float* xn    = W + o; o += (size_t)SEQ * D_MODEL;
    float* xz    = W + o; o += (size_t)SEQ * 2 * ED;
    float* xc    = W + o; o += (size_t)SEQ * ED;
    float* dbc   = W + o; o += (size_t)SEQ * DBC_LD;
    float* delta = W + o; o += (size_t)SEQ * ED;
    float* yg    = W + o; o += (size_t)SEQ * ED;
    float* ap    = W + o; o += (size_t)CHUNKS * ED * D_STATE;
    float* be    = W + o; o += (size_t)CHUNKS * ED * D_STATE;
    float* hin   = W + o; o += (size_t)CHUNKS * ED * D_STATE;

    for (int layer = 0; layer < N_LAYERS; ++layer) {
        const float* cur_x  = (layer == 0) ? x_in : out;
        const float* nw_l   = norm_w + (size_t)layer * D_MODEL;
        const float* Win_l  = Win    + (size_t)layer * 2 * ED * D_MODEL;
        const float* cw_l   = conv_w + (size_t)layer * ED * D_CONV;
        const float* cb_l   = conv_b + (size_t)layer * ED;
        const float* Wx_l   = Wx     + (size_t)layer * DBC_LD * ED;
        const float* Wdt_l  = Wdt    + (size_t)layer * ED * DT_RANK;
        const float* bdt_l  = bdt    + (size_t)layer * ED;
        const float* Alog_l = A_log  + (size_t)layer * ED * D_STATE;
        const float* Dp_l   = Dp     + (size_t)layer * ED;
        const float* Wout_l = Wout   + (size_t)layer * D_MODEL * ED;

        // 1) RMSNorm
        rmsnorm_kernel<<<SEQ, 256, 0, stream>>>(cur_x, nw_l, xn);

        // 2) in_proj: xz[2048][4096] = xn x Win^T   (M=2048,N=4096,K=1024)
        {
            dim3 grid((2 * ED) / 64, SEQ / 32);
            gemm_wmma_kernel<2, 4, 0><<<grid, 32, 0, stream>>>(
                xn, D_MODEL, Win_l, D_MODEL, xz, 2 * ED, D_MODEL, nullptr, nullptr);
        }

        // 3) depthwise causal conv + SiLU
        conv_silu_kernel<<<(SEQ * ED) / 256, 256, 0, stream>>>(xz, cw_l, cb_l, xc);

        // 4) x_proj: dbc[2048][96] = xc x Wx^T   (N=96 = 6*16)
        {
            dim3 grid(1, SEQ / 32);
            gemm_wmma_kernel<2, 6, 0><<<grid, 32, 0, stream>>>(
                xc, ED, Wx_l, ED, dbc, DBC_LD, ED, nullptr, nullptr);
        }

        // 5) dt_proj + softplus: delta[2048][2048] = dbc[:, :64] x Wdt^T
        {
            dim3 grid(ED / 64, SEQ / 32);
            gemm_wmma_kernel<2, 4, 1><<<grid, 32, 0, stream>>>(
                dbc, DBC_LD, Wdt_l, DT_RANK, delta, ED, DT_RANK, bdt_l, nullptr);
        }

        // 6-8) chunked selective scan
        scan_pass1<<<(CHUNKS * ED) / 256, 256, 0, stream>>>(delta, xc, dbc, Alog_l, ap, be);
        scan_pass2<<<(ED * D_STATE) / 256, 256, 0, stream>>>(ap, be, hin);
        scan_pass3<<<(CHUNKS * ED) / 256, 256, 0, stream>>>(delta, xc, dbc, Alog_l,
                                                            Dp_l, xz, hin, yg);

        // 9) out_proj + residual: out[2048][1024] = res + yg x Wout^T
        {
            dim3 grid(D_MODEL / 64, SEQ / 32);
            const float* res = (layer == 0) ? x_in : out;
            gemm_wmma_kernel<2, 4, 2><<<grid, 32, 0, stream>>>(
                yg, ED, Wout_l, ED, out, D_MODEL, ED, nullptr, res);
        }
    }
}